// EBGANGeneratorLoss_78159814853092
// MI455X (gfx1250) — compile-verified
//
#include <hip/hip_runtime.h>

typedef __attribute__((ext_vector_type(2))) float v2f;
typedef __attribute__((ext_vector_type(4))) float v4f;
typedef __attribute__((ext_vector_type(8))) float v8f;

static __device__ __forceinline__ float waveReduceSum(float v) {
  // wave32 butterfly reduction
  for (int m = 16; m >= 1; m >>= 1) v += __shfl_xor(v, m, 32);
  return v;
}

static __device__ __forceinline__ v8f wmma_accum(v2f a, v2f b, v8f c) {
  return __builtin_amdgcn_wmma_f32_16x16x4_f32(
      false, a, false, b, (short)0, c, false, false);
}

// ---------------------------------------------------------------------------
// Kernel A: MSE partial sums. Each lane streams float4 (global_load_b128) from
// both arrays; squared diffs are folded through v_wmma_f32_16x16x4_f32 with
// B = all-ones (D[m,n] += row-sums of A, replicated across 16 columns ->
// final sum carries a 16x factor; corrected in finalize).
// ---------------------------------------------------------------------------
__global__ void __launch_bounds__(256) mse_partial_kernel(
    const float* __restrict__ yhat, const float* __restrict__ xhat,
    float* __restrict__ partials, int chunksPerWave) {
  __shared__ float red[8];
  const int tid  = threadIdx.x;
  const int lane = tid & 31;
  const int wib  = tid >> 5;
  const long long waveId = (long long)blockIdx.x * 8 + wib;
  // Each chunk = 128 consecutive floats; lane covers [lane*4, lane*4+3].
  const long long base = waveId * (long long)chunksPerWave * 128LL + lane * 4;

  v8f acc = {};
  const v2f ones = {1.0f, 1.0f};
  for (int c = 0; c < chunksPerWave; ++c) {
    long long i = base + (long long)c * 128;
    v4f y = *(const v4f*)(yhat + i);
    v4f x = *(const v4f*)(xhat + i);
    float d0 = y.x - x.x;
    float d1 = y.y - x.y;
    float d2 = y.z - x.z;
    float d3 = y.w - x.w;
    v2f a0; a0.x = d0 * d0; a0.y = d1 * d1;
    v2f a1; a1.x = d2 * d2; a1.y = d3 * d3;
    acc = wmma_accum(a0, ones, acc);
    acc = wmma_accum(a1, ones, acc);
  }
  float local = 0.f;
  for (int r = 0; r < 8; ++r) local += acc[r];
  local = waveReduceSum(local);
  if (lane == 0) red[wib] = local;
  __syncthreads();
  if (tid == 0) {
    float s = 0.f;
    for (int i = 0; i < 8; ++i) s += red[i];
    partials[blockIdx.x] = s;  // carries 16x replication factor
  }
}

// ---------------------------------------------------------------------------
// Kernel B: per-row z term via closed form sum_{i!=j}(ẑ_i ẑ_j)^2 = 1 - s4/s2^2.
// One wave per 16 rows. A-matrix layout (16x4 f32): lanes 0-15 supply
// (M=lane, K=0..1), lanes 16-31 supply (M=lane-16, K=2..3). Each lane loads a
// float4: lanes 0-15 cover columns 8k..8k+3, lanes 16-31 cover 8k+4..8k+7 of
// the same rows; with B = ones the K mapping is immaterial, D[m,*] = s2 / s4
// per row (replicated 16x across columns).
// ---------------------------------------------------------------------------
__global__ void __launch_bounds__(32) zrow_kernel(
    const float* __restrict__ z, float* __restrict__ partials, int h) {
  const int lane = threadIdx.x & 31;
  const int tile = blockIdx.x;
  const int rowInTile = lane & 15;
  const int kOff = (lane >> 4) << 2;  // 0 for lanes 0-15, 4 for lanes 16-31
  const int row = tile * 16 + rowInTile;
  const float* zr = z + (long long)row * h + kOff;

  v8f c2 = {}, c4 = {};
  const v2f ones = {1.0f, 1.0f};
  const int kChunks = h >> 3;  // 8 columns per wave-chunk
  for (int kc = 0; kc < kChunks; ++kc) {
    v4f zv = *(const v4f*)(zr + kc * 8);
    float q0 = zv.x * zv.x;
    float q1 = zv.y * zv.y;
    float q2 = zv.z * zv.z;
    float q3 = zv.w * zv.w;
    v2f a2_0; a2_0.x = q0;      a2_0.y = q1;
    v2f a2_1; a2_1.x = q2;      a2_1.y = q3;
    v2f a4_0; a4_0.x = q0 * q0; a4_0.y = q1 * q1;
    v2f a4_1; a4_1.x = q2 * q2; a4_1.y = q3 * q3;
    c2 = wmma_accum(a2_0, ones, c2);
    c2 = wmma_accum(a2_1, ones, c2);
    c4 = wmma_accum(a4_0, ones, c4);
    c4 = wmma_accum(a4_1, ones, c4);
  }
  // C/D layout: VGPR r, lanes 0-15 -> M=r; lanes 16-31 -> M=r+8. Every row
  // appears in 16 lanes, so the wave-sum carries a 16x factor.
  float local = 0.f;
  for (int r = 0; r < 8; ++r) {
    float s2 = c2[r];
    float s4 = c4[r];
    local += 1.0f - s4 / (s2 * s2);
  }
  local = waveReduceSum(local);
  if (lane == 0) partials[tile] = local;  // 16 * (sum of 16 row terms)
}

// ---------------------------------------------------------------------------
// Kernel C: fold partials + any scalar tail into the final loss value.
// ---------------------------------------------------------------------------
__global__ void finalize_kernel(const float* __restrict__ msePartials, int nMse,
                                const float* __restrict__ zPartials, int nZ,
                                const float* __restrict__ yhat,
                                const float* __restrict__ xhat,
                                long long tailStart, long long N,
                                float* __restrict__ out,
                                float lambda, float invBHH) {
  if (threadIdx.x != 0 || blockIdx.x != 0) return;
  float s = 0.f;
  for (int i = 0; i < nMse; ++i) s += msePartials[i];
  s *= (1.0f / 16.0f);  // undo WMMA column replication
  for (long long i = tailStart; i < N; ++i) {  // empty when N divides evenly
    float d = yhat[i] - xhat[i];
    s += d * d;
  }
  float mse = s / (float)N;

  float zs = 0.f;
  for (int i = 0; i < nZ; ++i) zs += zPartials[i];
  zs *= (1.0f / 16.0f);  // undo replication

  out[0] = mse + lambda * zs * invBHH;
}

extern "C" void kernel_launch(void* const* d_in, const int* in_sizes, int n_in,
                              void* d_out, int out_size, void* d_ws, size_t ws_size,
                              hipStream_t stream) {
  const float* yhat = (const float*)d_in[0];
  const float* xhat = (const float*)d_in[1];
  const float* z    = (const float*)d_in[2];
  float* out = (float*)d_out;
  float* ws  = (float*)d_ws;

  const long long N = (long long)in_sizes[0];  // 3,145,728
  const int h = 512;
  const int b = in_sizes[2] / h;               // 256

  const int mseBlocks = 256, mseThreads = 256;
  const long long totalWaves = (long long)mseBlocks * (mseThreads / 32);  // 2048
  const int chunksPerWave = (int)(N / (128LL * totalWaves));              // 12
  const long long tailStart = (long long)chunksPerWave * 128LL * totalWaves;

  float* msePartials = ws;                 // [256]
  float* zPartials   = ws + mseBlocks;     // [16]
  const int zTiles = b / 16;               // 16

  mse_partial_kernel<<<mseBlocks, mseThreads, 0, stream>>>(
      yhat, xhat, msePartials, chunksPerWave);
  zrow_kernel<<<zTiles, 32, 0, stream>>>(z, zPartials, h);
  finalize_kernel<<<1, 32, 0, stream>>>(
      msePartials, mseBlocks, zPartials, zTiles,
      yhat, xhat, tailStart, N, out,
      0.1f, 1.0f / ((float)b * (float)h * (float)h));
}